// FBP_4002909520507
// MI455X (gfx1250) — compile-verified
//
#include <hip/hip_runtime.h>

// ---------------- geometry constants (match reference) ----------------
#define VIEWS 160
#define DETS  640
#define IMG_H 416
#define IMG_W 416
#define BATCH 32

__device__ constexpr float D_IMG   = 0.006641f;
__device__ constexpr float D_DET   = 0.012858f;
__device__ constexpr float D_ANG   = 0.009817477f * 4.0f;
__device__ constexpr float S2R     = 5.95f;
__device__ constexpr float D2R     = 4.906f;
__device__ constexpr float VIRDET  = D_DET * S2R / (S2R + D2R);
__device__ constexpr float PI_F    = 3.14159265358979323846f;
__device__ constexpr float COEF    = PI_F / (float)VIEWS;

typedef float v2f __attribute__((ext_vector_type(2)));
typedef float v8f __attribute__((ext_vector_type(8)));

// ---------------------------------------------------------------------
// Kernel 1: FwT[j][i] = w[i] * filt[i - j + (DETS-1)]   (K-major layout)
// Cosine pre-weighting folded into the ramp-filter matrix; stored
// transposed so the GEMM's B-operand is K-contiguous (b64 loads).
// ---------------------------------------------------------------------
__global__ void build_fw_kernel(const float* __restrict__ w,
                                const float* __restrict__ filt,
                                float* __restrict__ FwT) {
    int t = blockIdx.x * blockDim.x + threadIdx.x;   // t = j*DETS + i
    if (t >= DETS * DETS) return;
    int j = t / DETS;
    int i = t - j * DETS;
    FwT[t] = w[i] * filt[i - j + (DETS - 1)];
}

// ---------------------------------------------------------------------
// Kernel 2: Q(5120 x 640) = P(5120 x 640) @ Fw(640 x 640), fp32 WMMA.
// One wave per 16x16 output tile, K-loop in steps of 4 using
// V_WMMA_F32_16X16X4_F32.  8 waves (256 threads) per block.
//
// A 16x4 f32 layout (ISA 7.12.2): lanes 0-15 hold M=lane, K={k0,k0+1};
// lanes 16-31 hold M=lane-16, K={k0+2,k0+3}.  B 4x16 mirrors with the
// N index striped across lanes.  C/D: VGPR v -> M = v + 8*(lane>=16),
// N = lane&15.  Both operands are K-contiguous -> one b64 load each.
// ---------------------------------------------------------------------
__global__ void __launch_bounds__(256)
filter_gemm_kernel(const float* __restrict__ P,
                   const float* __restrict__ FwT,
                   float* __restrict__ Q) {
    const int lane = threadIdx.x & 31;
    const int wave = threadIdx.x >> 5;
    const int tile = blockIdx.x * 8 + wave;          // 12800 tiles total
    const int tM   = tile / (DETS / 16);             // 0..319
    const int tN   = tile - tM * (DETS / 16);        // 0..39
    const int half = lane >> 4;                      // 0 or 1
    const int l15  = lane & 15;

    const float* __restrict__ pa = P   + (size_t)(tM * 16 + l15) * DETS;
    const float* __restrict__ pb = FwT + (size_t)(tN * 16 + l15) * DETS;

    v8f acc = {};
    for (int k0 = 0; k0 < DETS; k0 += 4) {
        const int kk = k0 + 2 * half;
        v2f a = *(const v2f*)(pa + kk);
        v2f b = *(const v2f*)(pb + kk);
        acc = __builtin_amdgcn_wmma_f32_16x16x4_f32(
            /*neg_a=*/false, a, /*neg_b=*/false, b,
            /*c_mod=*/(short)0, acc, /*reuse_a=*/false, /*reuse_b=*/false);
    }

    const int rBase = tM * 16 + 8 * half;
#pragma unroll
    for (int v = 0; v < 8; ++v)
        Q[(size_t)(rBase + v) * DETS + tN * 16 + l15] = acc[v];
}

// ---------------------------------------------------------------------
// Kernel 3: fan-beam backprojection.
// 512 threads = 16 wave32s; each thread owns one pixel for 16 batches
// at a time (two batch groups).  Per view, 16 filtered sinogram rows
// (16 x 640 f32 = 40 KB) are staged in LDS via CDNA5 async
// global->LDS B128 copies (5 instructions/thread, ASYNCcnt tracked).
// The LDS destination operand is derived from &sh[0] (ptrtoint) so the
// array is captured by the asm and the subsequent ds_loads are NOT
// eliminated by alias analysis; the low 32 bits of a generic LDS
// pointer are exactly the workgroup-relative byte address VDST wants.
// Per-pixel geometry (idx / frac / weight) is computed once per view
// and reused for all batches, so the inner loop is pure ds_load+v_fma.
// ---------------------------------------------------------------------
__global__ void __launch_bounds__(512)
backproject_kernel(const float* __restrict__ Q, float* __restrict__ out) {
    __shared__ float sh[16 * DETS];                  // 40 KB

    const int tid   = threadIdx.x;
    const int pixel = blockIdx.x * 512 + tid;        // 0..173055
    const int py    = pixel / IMG_W;
    const int px    = pixel - py * IMG_W;

    const float x = ((float)px - (IMG_W - 1) * 0.5f) * D_IMG;
    const float y = ((IMG_H - 1) * 0.5f - (float)py) * D_IMG;

    // Workgroup-relative LDS byte address of sh[0]; the ptrtoint makes
    // sh escape into the asm below (keeps the ds_loads alive).
    const unsigned lbase = (unsigned)(size_t)(void*)sh;

    for (int g = 0; g < 2; ++g) {                    // batch groups of 16
        float acc[16];
#pragma unroll
        for (int b = 0; b < 16; ++b) acc[b] = 0.0f;

        for (int v = 0; v < VIEWS; ++v) {
            // async LDS fill: 16 rows x 640 f32 = 2560 16-byte chunks,
            // 512 lanes x 5 chunks, global_load_async_to_lds_b128
#pragma unroll
            for (int c = 0; c < 5; ++c) {
                const int ci = tid + c * 512;        // chunk id 0..2559
                const int b  = ci / 160;             // 160 chunks per row
                const int d4 = ci - b * 160;
                const float* gp = Q + (size_t)((g * 16 + b) * VIEWS + v) * DETS
                                    + d4 * 4;
                const unsigned ldsaddr = lbase + (unsigned)(ci * 16);
                asm volatile("global_load_async_to_lds_b128 %0, %1, off"
                             :: "v"(ldsaddr), "v"(gp)
                             : "memory");
            }
            asm volatile("s_wait_asynccnt 0x0" ::: "memory");
            __syncthreads();

            float sb, cb;
            __sincosf((float)v * D_ANG, &sb, &cb);
            const float d   = x * cb + y * sb;
            const float r   = -x * sb + y * cb;
            const float iU  = 1.0f / (S2R - d);      // U > 0 always here
            const float idx = (S2R * r * iU) * (1.0f / VIRDET)
                              + (DETS - 1) * 0.5f;
            const bool valid = (idx >= 0.0f) && (idx <= (float)(DETS - 1));
            float fi0 = floorf(idx);
            fi0 = fminf(fmaxf(fi0, 0.0f), (float)(DETS - 2));
            const int   i0   = (int)fi0;
            const float frac = fminf(fmaxf(idx - fi0, 0.0f), 1.0f);
            const float su   = S2R * iU;
            const float wgt  = valid ? su * su : 0.0f;

#pragma unroll
            for (int b = 0; b < 16; ++b) {
                const float v0 = sh[b * DETS + i0];
                const float v1 = sh[b * DETS + i0 + 1];
                acc[b] += wgt * (v0 + frac * (v1 - v0));
            }
            __syncthreads();
        }

#pragma unroll
        for (int b = 0; b < 16; ++b)
            out[(size_t)(g * 16 + b) * (IMG_H * IMG_W) + pixel] = COEF * acc[b];
    }
}

// ---------------------------------------------------------------------
extern "C" void kernel_launch(void* const* d_in, const int* in_sizes, int n_in,
                              void* d_out, int out_size, void* d_ws, size_t ws_size,
                              hipStream_t stream) {
    const float* proj = (const float*)d_in[0];   // [32,1,160,640]
    const float* w    = (const float*)d_in[1];   // [1,1,1,640]
    const float* filt = (const float*)d_in[2];   // [1,1,1,1279]
    float*       out  = (float*)d_out;           // [32,1,416,416]

    float* FwT = (float*)d_ws;                   // 640*640 f32 (1.6 MB)
    float* Q   = FwT + DETS * DETS;              // 5120*640 f32 (13.1 MB)

    // 1) weighted ramp-filter matrix (transposed / K-major)
    build_fw_kernel<<<(DETS * DETS + 255) / 256, 256, 0, stream>>>(w, filt, FwT);

    // 2) filtering as fp32 WMMA GEMM: 320 x 40 tiles, 8 waves/block
    const int tiles = (BATCH * VIEWS / 16) * (DETS / 16);   // 12800
    filter_gemm_kernel<<<tiles / 8, 256, 0, stream>>>(proj, FwT, Q);

    // 3) backprojection: 173056 pixels / 512 per block = 338 blocks
    backproject_kernel<<<(IMG_H * IMG_W) / 512, 512, 0, stream>>>(Q, out);
}